// Model_85169201480059
// MI455X (gfx1250) — compile-verified
//
#include <hip/hip_runtime.h>
#include <hip/hip_bf16.h>
#include <math.h>

#define BSZ 2048
#define LL 64
#define DD 15
#define NCLS 10
#define TREE_LEN 127

typedef __attribute__((ext_vector_type(2))) float v2f;
typedef __attribute__((ext_vector_type(8))) float v8f;

// rms_norm over 15 components (matches reference: x / (sqrt(mean(x^2)+1e-6)+1e-6))
__device__ __forceinline__ void rms_norm15(float* v) {
  float s = 0.f;
#pragma unroll
  for (int d = 0; d < DD; ++d) s = fmaf(v[d], v[d], s);
  float inv = 1.0f / (sqrtf(s * (1.0f / 15.0f) + 1e-6f) + 1e-6f);
#pragma unroll
  for (int d = 0; d < DD; ++d) v[d] *= inv;
}

// combine(a,b): logits l_r = a . (M_r b), alpha = softmax([l0,l1]), out = a0*a + a1*b
// W is lane-uniform -> compiler emits scalar (s_load) reads, FMAs take SGPR operand.
__device__ __forceinline__ void combine15(const float* __restrict__ W,
                                          const float* a, const float* b, float* out) {
  float l0 = 0.f, l1 = 0.f;
#pragma unroll
  for (int i = 0; i < DD; ++i) {
    float s0 = 0.f, s1 = 0.f;
#pragma unroll
    for (int j = 0; j < DD; ++j) {
      s0 = fmaf(W[i * DD + j], b[j], s0);
      s1 = fmaf(W[DD * DD + i * DD + j], b[j], s1);
    }
    l0 = fmaf(a[i], s0, l0);
    l1 = fmaf(a[i], s1, l1);
  }
  float mx = fmaxf(l0, l1);
  float e0 = expf(l0 - mx);
  float e1 = expf(l1 - mx);
  float inv = 1.0f / (e0 + e1);
  float a0 = e0 * inv;
  float a1 = e1 * inv;
#pragma unroll
  for (int d = 0; d < DD; ++d) out[d] = fmaf(a0, a[d], a1 * b[d]);
}

// One sample per lane; per-sample 127x15 tree lives in LDS (CDNA5: 320KB/WGP).
// Layout [node][d][lane] -> every ds access is bank-conflict-free across the wave.
__global__ __launch_bounds__(32, 1) void seg_tree_kernel(
    const float* __restrict__ emb, const float* __restrict__ W_rule,
    const int* __restrict__ x, const int* __restrict__ qlo,
    const int* __restrict__ qhi, float* __restrict__ ws_ret) {
  __shared__ float s_tree[TREE_LEN * DD * 32];  // 243,840 B
  const int lane = threadIdx.x;
  const int s = blockIdx.x * 32 + lane;
  const int ql = qlo[s];
  const int qh = qhi[s];

  float inf_raw[DD];  // emb[NC], un-normalized (lane-uniform -> scalar loads)
#pragma unroll
  for (int d = 0; d < DD; ++d) inf_raw[d] = emb[NCLS * DD + d];

  // ---------------- build: leaves (nodes 63..126, leaf i at node 63+i) ----------------
  for (int i = 0; i < LL; ++i) {
    const int c = x[s * LL + i];
    float v[DD];
#pragma unroll
    for (int d = 0; d < DD; ++d) v[d] = emb[c * DD + d];
    rms_norm15(v);
    rms_norm15(v);  // repeated whole-tree norms in the reference converge after 2 apps
    const int p = 63 + i;
#pragma unroll
    for (int d = 0; d < DD; ++d) s_tree[(p * DD + d) * 32 + lane] = v[d];
  }
  // ---------------- build: internal nodes (post-order == descending index) ------------
  for (int p = 62; p >= 0; --p) {
    float a[DD], b[DD], v[DD];
#pragma unroll
    for (int d = 0; d < DD; ++d) {
      a[d] = s_tree[((2 * p + 1) * DD + d) * 32 + lane];
      b[d] = s_tree[((2 * p + 2) * DD + d) * 32 + lane];
    }
    combine15(W_rule, a, b, v);
    rms_norm15(v);
    rms_norm15(v);
#pragma unroll
    for (int d = 0; d < DD; ++d) s_tree[(p * DD + d) * 32 + lane] = v[d];
  }

  // ---------------- query pass 1: top-down, write r2[p] in place ----------------------
  // r2[p] = mB ? inf_raw : (mA ? tree[p] : r2[parent])  (masks are 0/1 -> exact selects)
  for (int p = 0; p < TREE_LEN; ++p) {
    const int t = p + 1;
    const int dep = 31 - __clz(t);
    const int seg = LL >> dep;
    const int lo = (t - (1 << dep)) * seg;
    const int hi = lo + seg - 1;
    const bool mA = (ql <= lo) && (qh >= hi);
    const bool mB = (ql > hi) || (qh < lo);
    const int par = (p - 1) >> 1;
#pragma unroll
    for (int d = 0; d < DD; ++d) {
      const float tv = s_tree[(p * DD + d) * 32 + lane];
      const float pv = (p == 0) ? 0.f : s_tree[(par * DD + d) * 32 + lane];
      const float r1 = mA ? tv : pv;
      s_tree[(p * DD + d) * 32 + lane] = mB ? inf_raw[d] : r1;
    }
  }

  // ---------------- query pass 2: bottom-up, out[p] overwrites r2[p] ------------------
  // out[p] = rms_norm(mC * combine(out[l], out[r]) + (1-mC) * r2[p]); leaf children == r2
  for (int p = TREE_LEN - 1; p >= 0; --p) {
    const int t = p + 1;
    const int dep = 31 - __clz(t);
    const int seg = LL >> dep;
    const int lo = (t - (1 << dep)) * seg;
    const int hi = lo + seg - 1;
    const bool mA = (ql <= lo) && (qh >= hi);
    const bool mB = (ql > hi) || (qh < lo);
    const bool mC = (!mA) && (!mB);
    float r2[DD], c1[DD], c2[DD];
#pragma unroll
    for (int d = 0; d < DD; ++d) r2[d] = s_tree[(p * DD + d) * 32 + lane];
    if (p < 63) {
#pragma unroll
      for (int d = 0; d < DD; ++d) {
        c1[d] = s_tree[((2 * p + 1) * DD + d) * 32 + lane];
        c2[d] = s_tree[((2 * p + 2) * DD + d) * 32 + lane];
      }
    } else {
#pragma unroll
      for (int d = 0; d < DD; ++d) { c1[d] = r2[d]; c2[d] = r2[d]; }
    }
    float nv[DD], o[DD];
    combine15(W_rule, c1, c2, nv);
#pragma unroll
    for (int d = 0; d < DD; ++d) o[d] = mC ? nv[d] : r2[d];
    rms_norm15(o);
#pragma unroll
    for (int d = 0; d < DD; ++d) s_tree[(p * DD + d) * 32 + lane] = o[d];
  }

  // ---------------- stage returns, K-padded to 16 for the WMMA projection -------------
#pragma unroll
  for (int d = 0; d < DD; ++d) ws_ret[s * 16 + d] = s_tree[d * 32 + lane];
  ws_ret[s * 16 + DD] = 0.f;
}

// Final projection: out(2048x10) = returns(2048x15) @ W_lin^T, via V_WMMA_F32_16X16X4_F32.
// One wave per 16-sample tile; K=16 (padded) -> 4 WMMA k-steps; N padded 10->16.
// f32 A 16x4 layout: vgpr0 -> K = 2*(lane/16), vgpr1 -> K = 2*(lane/16)+1, M = lane%16.
// B 4x16 mirror with N = lane%16. C/D: vgpr r -> M = r + 8*(lane/16), N = lane%16.
// W_lin is staged zero-padded in LDS so B fetches are unconditional ds_load_b64
// (no exec-mask predication; EXEC stays all-ones as WMMA requires).
__global__ __launch_bounds__(256) void proj_kernel(const float* __restrict__ ws_ret,
                                                   const float* __restrict__ W_lin,
                                                   float* __restrict__ out) {
  __shared__ float sW[16 * 16];  // sW[n*16 + k] = W_lin^T padded (n<10, k<15 real)
  {
    const int idx = threadIdx.x;  // 256 threads fill 256 entries, no branches
    const int n = idx >> 4;
    const int k = idx & 15;
    const int cidx = (n < NCLS ? n : NCLS - 1) * DD + (k < DD ? k : DD - 1);
    const float w = W_lin[cidx];                       // clamped, unconditional load
    sW[idx] = (n < NCLS && k < DD) ? w : 0.f;          // select after load
  }
  __syncthreads();

  const int lane = threadIdx.x & 31;
  const int tile = blockIdx.x * 8 + (threadIdx.x >> 5);
  const int m = lane & 15;         // A row / B col / D col index
  const int half = lane >> 4;      // lane half selects K pair

  v8f acc = {};
#pragma unroll
  for (int step = 0; step < 4; ++step) {
    const int k0 = step * 4 + 2 * half;
    v2f A, B;
    A.x = ws_ret[(tile * 16 + m) * 16 + k0];
    A.y = ws_ret[(tile * 16 + m) * 16 + k0 + 1];
    B.x = sW[m * 16 + k0];
    B.y = sW[m * 16 + k0 + 1];
    acc = __builtin_amdgcn_wmma_f32_16x16x4_f32(false, A, false, B, (short)0, acc,
                                                false, false);
  }
  if (m < NCLS) {
#pragma unroll
    for (int r = 0; r < 8; ++r) {
      const int mm = r + 8 * half;
      out[(tile * 16 + mm) * NCLS + m] = acc[r];
    }
  }
}

extern "C" void kernel_launch(void* const* d_in, const int* in_sizes, int n_in,
                              void* d_out, int out_size, void* d_ws, size_t ws_size,
                              hipStream_t stream) {
  const float* emb = (const float*)d_in[0];
  const float* W_lin = (const float*)d_in[1];
  const float* W_rule = (const float*)d_in[2];
  const int* x = (const int*)d_in[3];
  const int* qlo = (const int*)d_in[4];
  const int* qhi = (const int*)d_in[5];
  float* out = (float*)d_out;
  float* ws_ret = (float*)d_ws;  // 2048 * 16 floats = 128 KB staging

  seg_tree_kernel<<<BSZ / 32, 32, 0, stream>>>(emb, W_rule, x, qlo, qhi, ws_ret);
  proj_kernel<<<(BSZ / 16) / 8, 256, 0, stream>>>(ws_ret, W_lin, out);
}